// MACD_RNN_69363721830612
// MI455X (gfx1250) — compile-verified
//
#include <hip/hip_runtime.h>
#include <hip/hip_bf16.h>

typedef __attribute__((ext_vector_type(2))) float v2f;
typedef __attribute__((ext_vector_type(8))) float v8f;

// MACD over [B=4096, T=2048, F=8] f32.
//   y_slow = EMA(N=26) -> alpha = 2/27, beta = 25/27
//   y_fast = EMA(N=12) -> alpha = 2/13, beta = 11/13
//   out = y_slow - y_fast
//
// Block-scan formulation: for a 16-step chunk with incoming carry c,
//   y_i = beta^(i+1) * c + sum_{j<=i} alpha*beta^(i-j) * x_j
// => Y(16x16) = M(16x16, const lower-tri) * X(16 time x 16 series) + d (x) c
// computed as 4 chained V_WMMA_F32_16X16X4_F32 per EMA with the carry term
// preloaded into the accumulator. New carry = row 15 of D (ds_bpermute).
//
// Traffic: 256 MB read + 256 MB write -> ~22 us floor at 23.3 TB/s. The WMMA
// formulation breaks the 2048-step serial scan into 128 carry hops per wave.

__global__ __launch_bounds__(256) void macd_wmma_kernel(
    const float* __restrict__ x, float* __restrict__ out)
{
    constexpr int   T  = 2048;
    constexpr int   F  = 8;
    constexpr float AS = 2.0f / 27.0f, BS = 25.0f / 27.0f;   // slow (N=26)
    constexpr float AF = 2.0f / 13.0f, BF = 11.0f / 13.0f;   // fast (N=12)

    const int tid  = blockIdx.x * blockDim.x + threadIdx.x;
    const int lane = threadIdx.x & 31;
    const int wave = tid >> 5;          // 0..2047, each wave owns 16 series
    const int n    = lane & 15;         // series column within the tile
    const int hi   = lane >> 4;         // lane half (0/1)

    // Series s = b*8 + f; this wave covers s = wave*16 .. wave*16+15
    const int  s    = wave * 16 + n;
    const long soff = (long)(s >> 3) * (long)(T * F) + (long)(s & 7);

    // Cheap integer-power coefficients: beta^e = exp2(e * log2(beta)).
    // v_log_f32 / v_exp_f32 via amdgcn builtins (HIP __exp2f is host-only here).
    const float l2s = __builtin_amdgcn_logf(BS);
    const float l2f = __builtin_amdgcn_logf(BF);

    // A-matrix chunks: A_k holds columns j = 4k..4k+3 of M.
    // 32-bit 16x4 A layout: component p of lane -> K = p + 2*hi, M = lane&15.
    const int m = n;
    v2f aS[4], aF[4];
    #pragma unroll
    for (int k = 0; k < 4; ++k) {
        #pragma unroll
        for (int p = 0; p < 2; ++p) {
            const int   j = 4 * k + p + 2 * hi;
            const float e = (float)(m - j);
            aS[k][p] = (j <= m) ? AS * __builtin_amdgcn_exp2f(e * l2s) : 0.0f;
            aF[k][p] = (j <= m) ? AF * __builtin_amdgcn_exp2f(e * l2f) : 0.0f;
        }
    }

    // Carry weights d_m = beta^(m+1); D/C layout: VGPR v -> M = v + 8*hi.
    float dS[8], dF[8];
    #pragma unroll
    for (int v = 0; v < 8; ++v) {
        const float ev = (float)(v + 8 * hi + 1);
        dS[v] = __builtin_amdgcn_exp2f(ev * l2s);
        dF[v] = __builtin_amdgcn_exp2f(ev * l2f);
    }

    // B layout (32-bit 4x16): component p of lane -> row K = p + 2*hi.
    // So chunk-k pair loads times t0 + {4k, 4k+1} + 2*hi.
    const float* lp = x   + soff + (long)(2 * hi) * F;
    float*       op = out + soff + (long)(8 * hi) * F;

    // Seed carry with x[t=0] so y_0 = alpha*x0 + beta*x0 = x0 (EMA init).
    float carry_s = x[soff];
    float carry_f = carry_s;
    const int bidx = (16 + n) * 4;   // bpermute byte index: row 15 lives in lanes 16..31

    for (int c = 0; c < T / 16; ++c) {
        v2f b0, b1, b2, b3;
        b0[0] = lp[0 * F];  b0[1] = lp[1 * F];
        b1[0] = lp[4 * F];  b1[1] = lp[5 * F];
        b2[0] = lp[8 * F];  b2[1] = lp[9 * F];
        b3[0] = lp[12 * F]; b3[1] = lp[13 * F];
        __builtin_prefetch(lp + 16 * F, 0, 3);   // global_prefetch_b8, next chunk

        v8f accS, accF;
        #pragma unroll
        for (int v = 0; v < 8; ++v) {
            accS[v] = dS[v] * carry_s;
            accF[v] = dF[v] * carry_f;
        }

        accS = __builtin_amdgcn_wmma_f32_16x16x4_f32(false, aS[0], false, b0, (short)0, accS, false, false);
        accS = __builtin_amdgcn_wmma_f32_16x16x4_f32(false, aS[1], false, b1, (short)0, accS, false, false);
        accS = __builtin_amdgcn_wmma_f32_16x16x4_f32(false, aS[2], false, b2, (short)0, accS, false, false);
        accS = __builtin_amdgcn_wmma_f32_16x16x4_f32(false, aS[3], false, b3, (short)0, accS, false, false);

        accF = __builtin_amdgcn_wmma_f32_16x16x4_f32(false, aF[0], false, b0, (short)0, accF, false, false);
        accF = __builtin_amdgcn_wmma_f32_16x16x4_f32(false, aF[1], false, b1, (short)0, accF, false, false);
        accF = __builtin_amdgcn_wmma_f32_16x16x4_f32(false, aF[2], false, b2, (short)0, accF, false, false);
        accF = __builtin_amdgcn_wmma_f32_16x16x4_f32(false, aF[3], false, b3, (short)0, accF, false, false);

        // New carries: D row 15 = VGPR7 of lanes 16..31, broadcast to both halves.
        carry_s = __int_as_float(__builtin_amdgcn_ds_bpermute(bidx, __float_as_int(accS[7])));
        carry_f = __int_as_float(__builtin_amdgcn_ds_bpermute(bidx, __float_as_int(accF[7])));

        // out = slow - fast; VGPR v -> time t0 + v + 8*hi (base already has 8*hi).
        #pragma unroll
        for (int v = 0; v < 8; ++v)
            op[v * F] = accS[v] - accF[v];

        lp += 16 * F;
        op += 16 * F;
    }
}

extern "C" void kernel_launch(void* const* d_in, const int* in_sizes, int n_in,
                              void* d_out, int out_size, void* d_ws, size_t ws_size,
                              hipStream_t stream) {
    const float* x = (const float*)d_in[0];
    float*     out = (float*)d_out;

    // 4096*8 = 32768 series, 16 per wave -> 2048 waves -> 65536 threads.
    const int threads = 256;                 // 8 wave32 per block
    const int blocks  = 65536 / threads;     // 256 blocks
    hipLaunchKernelGGL(macd_wmma_kernel, dim3(blocks), dim3(threads), 0, stream,
                       x, out);
}